// LabelRenumber_17403207483472
// MI455X (gfx1250) — compile-verified
//
#include <hip/hip_runtime.h>

#define NCLS 16
#define TPB  256
#define KUNR 4   // B128 chunks per thread per tile -> 2KB in flight per wave

#if defined(__gfx1250__)
#if __has_builtin(__builtin_amdgcn_global_load_async_to_lds_b128) && \
    __has_builtin(__builtin_amdgcn_global_store_async_from_lds_b128) && \
    __has_builtin(__builtin_amdgcn_s_wait_asynccnt)
#define HAVE_ASYNC_LDS 1
#endif
#endif

#if defined(HAVE_ASYNC_LDS)
// Exact parameter types per the hipcc diagnostic: int4-vector pointers,
// addrspace(1) for the global side, addrspace(3) for the LDS side.
typedef int v4i_t __attribute__((ext_vector_type(4)));
typedef v4i_t __attribute__((address_space(1))) *gv4i_p;
typedef v4i_t __attribute__((address_space(3))) *lv4i_p;
#endif

// ---------------------------------------------------------------------------
// 1a) Classic streaming copy: tiles of KUNR strided B128 loads, then KUNR
//     B128 stores. Each load instruction is 32 lanes x 16B = 512B contiguous;
//     4 independent loads in flight per wave before the waitcnt.
// ---------------------------------------------------------------------------
__global__ void img_copy_b128(const float4* __restrict__ src,
                              float4* __restrict__ dst,
                              long long n4) {
    const long long tile = (long long)TPB * KUNR;
    long long base = (long long)blockIdx.x * tile + threadIdx.x;
    long long step = (long long)gridDim.x * tile;

    for (; base < n4; base += step) {
#if defined(__gfx1250__)
        // speculative prefetch of next tile (global_prefetch_b8); dropped on
        // translation failure, so no bounds check needed
        __builtin_prefetch((const void*)&src[base + step], 0, 1);
#endif
        float4 r[KUNR];
        long long idx[KUNR];
#pragma unroll
        for (int k = 0; k < KUNR; ++k) {
            idx[k] = base + (long long)k * TPB;
            if (idx[k] < n4) r[k] = src[idx[k]];      // 4x global_load_b128
        }
#pragma unroll
        for (int k = 0; k < KUNR; ++k) {
            if (idx[k] < n4) dst[idx[k]] = r[k];      // 4x global_store_b128
        }
    }
}

// scalar tail copy (only launched when n_img % 4 != 0; not for this shape)
__global__ void img_copy_tail(const float* __restrict__ src,
                              float* __restrict__ dst,
                              long long start, long long n) {
    long long i = start + (long long)blockIdx.x * blockDim.x + threadIdx.x;
    if (i < n) dst[i] = src[i];
}

// ---------------------------------------------------------------------------
// 1b) Async global->LDS->global copy (CDNA5 async engine, ASYNCcnt).
//     Each lane owns its private 16B LDS slot per chunk, so only
//     s_wait_asynccnt ordering is needed (no cross-lane barrier).
// ---------------------------------------------------------------------------
__global__ void img_copy_async_lds(const float4* __restrict__ src,
                                   float4* __restrict__ dst,
                                   long long start4, long long n4) {
    const long long tile = (long long)TPB * KUNR;
    long long base = start4 + (long long)blockIdx.x * tile + threadIdx.x;
    long long step = (long long)gridDim.x * tile;
#if defined(HAVE_ASYNC_LDS)
    __shared__ float4 buf[TPB * KUNR];     // 16 KB of the 320 KB/WGP LDS
    unsigned t = threadIdx.x;
    for (; base < n4; base += step) {
#pragma unroll
        for (int k = 0; k < KUNR; ++k) {
            long long idx = base + (long long)k * TPB;
            if (idx < n4) {
                __builtin_amdgcn_global_load_async_to_lds_b128(
                    (gv4i_p)(float4*)&src[idx],
                    (lv4i_p)&buf[k * TPB + t],
                    0, 0);
            }
        }
        __builtin_amdgcn_s_wait_asynccnt(0);   // loads landed in LDS
#pragma unroll
        for (int k = 0; k < KUNR; ++k) {
            long long idx = base + (long long)k * TPB;
            if (idx < n4) {
                __builtin_amdgcn_global_store_async_from_lds_b128(
                    (gv4i_p)&dst[idx],
                    (lv4i_p)&buf[k * TPB + t],
                    0, 0);
            }
        }
        __builtin_amdgcn_s_wait_asynccnt(0);   // LDS slots free for reuse
    }
#else
    for (; base < n4; base += step) {
        float4 r[KUNR];
        long long idx[KUNR];
#pragma unroll
        for (int k = 0; k < KUNR; ++k) {
            idx[k] = base + (long long)k * TPB;
            if (idx[k] < n4) r[k] = src[idx[k]];
        }
#pragma unroll
        for (int k = 0; k < KUNR; ++k) {
            if (idx[k] < n4) dst[idx[k]] = r[k];
        }
    }
#endif
}

// ---------------------------------------------------------------------------
// 2) Re-init the 16-entry presence table in workspace each call.
// ---------------------------------------------------------------------------
__global__ void zero_presence(int* __restrict__ presence) {
    if (threadIdx.x < NCLS) presence[threadIdx.x] = 0;
}

// ---------------------------------------------------------------------------
// 3) Presence marking: LDS-first (idempotent "=1" stores, races benign),
//    then <=16 global stores per block.
// ---------------------------------------------------------------------------
__global__ void mark_presence(const int* __restrict__ label,
                              long long n,
                              int* __restrict__ presence) {
    __shared__ int lp[NCLS];
    if (threadIdx.x < NCLS) lp[threadIdx.x] = 0;
    __syncthreads();

    long long i      = (long long)blockIdx.x * blockDim.x + threadIdx.x;
    long long stride = (long long)gridDim.x * blockDim.x;
    long long n4     = n >> 2;
    const int4* l4   = (const int4*)label;

    for (long long j = i; j < n4; j += stride) {
        int4 v = l4[j];                       // global_load_b128
        lp[v.x & (NCLS - 1)] = 1;
        lp[v.y & (NCLS - 1)] = 1;
        lp[v.z & (NCLS - 1)] = 1;
        lp[v.w & (NCLS - 1)] = 1;
    }
    for (long long j = (n4 << 2) + i; j < n; j += stride)
        lp[label[j] & (NCLS - 1)] = 1;

    __syncthreads();
    if (threadIdx.x < NCLS && lp[threadIdx.x]) presence[threadIdx.x] = 1;
}

// ---------------------------------------------------------------------------
// 4) rank[v] = cumsum(presence)[v] - 1 built in LDS, then int4 gather ->
//    float4 store of rank[label].
// ---------------------------------------------------------------------------
__global__ void remap_labels(const int* __restrict__ label,
                             long long n,
                             const int* __restrict__ presence,
                             float* __restrict__ out) {
    __shared__ int rank[NCLS];
    if (threadIdx.x == 0) {
        int c = 0;
        for (int v = 0; v < NCLS; ++v) {
            c += presence[v];
            rank[v] = c - 1;
        }
    }
    __syncthreads();

    long long i      = (long long)blockIdx.x * blockDim.x + threadIdx.x;
    long long stride = (long long)gridDim.x * blockDim.x;
    long long n4     = n >> 2;
    const int4* l4   = (const int4*)label;
    float4*    o4    = (float4*)out;

    for (long long j = i; j < n4; j += stride) {
        int4 v = l4[j];
        float4 r;
        r.x = (float)rank[v.x & (NCLS - 1)];
        r.y = (float)rank[v.y & (NCLS - 1)];
        r.z = (float)rank[v.z & (NCLS - 1)];
        r.w = (float)rank[v.w & (NCLS - 1)];
        o4[j] = r;                            // global_store_b128
    }
    for (long long j = (n4 << 2) + i; j < n; j += stride)
        out[j] = (float)rank[label[j] & (NCLS - 1)];
}

// ---------------------------------------------------------------------------
extern "C" void kernel_launch(void* const* d_in, const int* in_sizes, int n_in,
                              void* d_out, int out_size, void* d_ws, size_t ws_size,
                              hipStream_t stream) {
    const float* img   = (const float*)d_in[0];
    const int*   label = (const int*)d_in[1];
    float*       out   = (float*)d_out;

    long long n_img = (long long)in_sizes[0];   // 4*103*512*512
    long long n_lab = (long long)in_sizes[1];   // 4*512*512
    int* presence = (int*)d_ws;                 // 16 ints of scratch

    const long long tile = (long long)TPB * KUNR;

    // --- image passthrough: float4-granular main body + scalar tail ---
    long long n4  = n_img >> 2;
    long long rem = n_img - (n4 << 2);

    // split at a tile boundary: first half classic B128, second half async-LDS
    long long split = (n4 >> 1) & ~(tile - 1);
    if (split < 0) split = 0;

    if (split > 0) {
        long long tiles = (split + tile - 1) / tile;
        int blocks = (tiles > 16384) ? 16384 : (tiles < 1 ? 1 : (int)tiles);
        img_copy_b128<<<blocks, TPB, 0, stream>>>((const float4*)img,
                                                  (float4*)out, split);
    }
    if (n4 - split > 0) {
        long long range = n4 - split;
        long long tiles = (range + tile - 1) / tile;
        int blocks = (tiles > 16384) ? 16384 : (tiles < 1 ? 1 : (int)tiles);
        img_copy_async_lds<<<blocks, TPB, 0, stream>>>((const float4*)img,
                                                       (float4*)out, split, n4);
    }
    if (rem > 0)
        img_copy_tail<<<1, TPB, 0, stream>>>(img, out, n4 << 2, n_img);

    // --- label renumber ---
    zero_presence<<<1, 32, 0, stream>>>(presence);

    long long lw = ((n_lab >> 2) + TPB - 1) / TPB;
    int lblocks = (lw > 2048) ? 2048 : (lw < 1 ? 1 : (int)lw);
    mark_presence<<<lblocks, TPB, 0, stream>>>(label, n_lab, presence);
    remap_labels<<<lblocks, TPB, 0, stream>>>(label, n_lab, presence, out + n_img);
}